// MemoryRNN_8589934761
// MI455X (gfx1250) — compile-verified
//
#include <hip/hip_runtime.h>
#include <math.h>

// ---------------------------------------------------------------------------
// MemoryRNN (test-time-training fast-weights) for MI455X / gfx1250, wave32.
//
//   B=8 independent samples, each: 64 strictly-sequential SGD steps on a
//   private 2-layer MLP (w1:1024x256, w2:256x1024) + momentum.
//
// Design:
//   * 1 workgroup (1024 threads = 32 wave32 waves) per sample, persistent.
//   * z = w1 @ xn via V_WMMA_F32_16X16X4_F32 (fp32 WMMA, exact precision).
//     w1 & its momentum live in d_ws pre-swizzled into the A-fragment lane
//     layout so each wave's A-tile load is one coalesced 256B global_load_b64.
//   * grads are rank-1 outer products -> never materialized; the global grad
//     norm is  sqrt(|dz|^2*|xn|^2 + |e|^2*|h|^2)  (4 tiny reductions).
//   * w2 stays row-major: pred (lane-split-K + shfl reduce), dh = w2^T e
//     (thread-per-column), and both update sweeps are fully coalesced.
// ---------------------------------------------------------------------------

namespace {
constexpr int kB = 8, kT = 64, kD = 256, kH = 1024;
constexpr float kLR = 1e-3f, kDecay = 0.02f, kMu = 0.9f, kEps = 1e-6f;
constexpr int kW1Elems = kH * kD;          // 262144
constexpr int kPerSample = 4 * kW1Elems;   // p1,m1,p2,m2
}

typedef float v2f __attribute__((ext_vector_type(2)));
typedef float v8f __attribute__((ext_vector_type(8)));

__device__ __forceinline__ float wave_reduce_add(float v) {
#pragma unroll
  for (int off = 16; off > 0; off >>= 1) v += __shfl_xor(v, off, 32);
  return v;
}

__device__ __forceinline__ float nan_to_num(float g) {
  if (g != g) return 0.0f;
  if (isinf(g)) return g > 0.0f ? 3.4028234663852886e38f : -3.4028234663852886e38f;
  return g;
}

__global__ __launch_bounds__(1024, 1) void memory_rnn_kernel(
    const float* __restrict__ keys,    // [B,T,D]
    const float* __restrict__ values,  // [B,T,D]
    const float* __restrict__ w1_in,   // [H,D]
    const float* __restrict__ w2_in,   // [D,H]
    float* __restrict__ out,           // preds [B,T,D] then surprises [B,T]
    float* __restrict__ ws) {
  const int b = blockIdx.x;
  const int tid = threadIdx.x;
  const int lane = tid & 31;
  const int wave = tid >> 5;

  float* p1 = ws + (size_t)b * kPerSample;          // WMMA-fragment-swizzled
  float* m1 = p1 + kW1Elems;                        // same swizzle
  float* p2 = m1 + kW1Elems;                        // row-major [D,H]
  float* m2 = p2 + kW1Elems;

  __shared__ float s_k[kD];
  __shared__ float s_xn[kD];
  __shared__ float s_v[kD];
  __shared__ float s_e[kD];
  __shared__ float s_z[kH];   // z, later overwritten with gelu'(z)
  __shared__ float s_h[kH];
  __shared__ float s_dz[kH];
  __shared__ float s_red[8];  // 0:ss 1:nrm 2:|xn|^2 3:|h|^2 4:|e|^2 5:|dz|^2 6:clip

  // ---- init: params from inputs (w1 swizzled to A-fragment order), moms=0 --
  // swizzled flat index: ((r*64 + ks)*32 + l)*2 + j
  //   row = 16r + (l&15),  d = 4ks + 2*(l>>4) + j
#pragma unroll 1
  for (int i = tid; i < kW1Elems; i += 1024) {
    int j = i & 1, l = (i >> 1) & 31, ks = (i >> 6) & 63, r = i >> 12;
    int hr = r * 16 + (l & 15);
    int d = ks * 4 + ((l >> 4) << 1) + j;
    p1[i] = w1_in[hr * kD + d];
    m1[i] = 0.0f;
    p2[i] = w2_in[i];
    m2[i] = 0.0f;
  }
  __threadfence();
  __syncthreads();

  const int koff = (lane >> 4) << 1;          // 0 or 2 (A/B K sub-offset)
  const bool bsel = ((lane & 15) == 0);       // only column N=0 of B is live

  for (int t = 0; t < kT; ++t) {
    // ---- Phase A: load k,v ; xn = k / max(||k||, 1e-12) ------------------
    if (tid < 8) s_red[tid] = 0.0f;
    if (tid < kD) s_k[tid] = keys[((size_t)b * kT + t) * kD + tid];
    else if (tid < 2 * kD) s_v[tid - kD] = values[((size_t)b * kT + t) * kD + (tid - kD)];
    __syncthreads();
    if (tid < kD) {
      float kv = s_k[tid];
      float p = wave_reduce_add(kv * kv);
      if (lane == 0) atomicAdd(&s_red[0], p);
    }
    __syncthreads();
    if (tid == 0) {
      float ss = s_red[0];
      float nrm = fmaxf(sqrtf(ss), 1e-12f);
      s_red[1] = nrm;
      s_red[2] = ss / (nrm * nrm);  // ||xn||^2 (exactly from the same fp path)
    }
    __syncthreads();
    if (tid < kD) s_xn[tid] = s_k[tid] * (1.0f / s_red[1]);
    __syncthreads();

    // ---- Phase B: z = w1 @ xn via V_WMMA_F32_16X16X4_F32 -----------------
    // 64 row-tiles of 16; each of 32 waves owns 2 tiles; K = 256 in 64 steps.
#pragma unroll 1
    for (int rep = 0; rep < 2; ++rep) {
      const int r = wave * 2 + rep;
      const float* abase = p1 + r * 4096;
      v8f c = {0.f, 0.f, 0.f, 0.f, 0.f, 0.f, 0.f, 0.f};
#pragma unroll 4
      for (int ks = 0; ks < 64; ++ks) {
        v2f a = *(const v2f*)(abase + ks * 64 + lane * 2);  // coalesced 256B/wave
        float x0 = s_xn[ks * 4 + koff];
        float x1 = s_xn[ks * 4 + koff + 1];
        v2f bb;
        bb.x = bsel ? x0 : 0.0f;   // B[K][N=0] = xn[K]; other columns zero
        bb.y = bsel ? x1 : 0.0f;
        c = __builtin_amdgcn_wmma_f32_16x16x4_f32(false, a, false, bb,
                                                  (short)0, c, false, false);
      }
      // C col 0: (M=v,N=0) in lane 0 vgpr v ; (M=v+8,N=0) in lane 16 vgpr v
      if (lane == 0) {
#pragma unroll
        for (int v = 0; v < 8; ++v) s_z[r * 16 + v] = c[v];
      } else if (lane == 16) {
#pragma unroll
        for (int v = 0; v < 8; ++v) s_z[r * 16 + 8 + v] = c[v];
      }
    }
    __syncthreads();

    // ---- Phase C: h = gelu(z) (exact); s_z := gelu'(z); accumulate |h|^2 --
    {
      float zz = s_z[tid];
      float cdf = 0.5f * (1.0f + erff(zz * 0.70710678118654752f));
      float pdf = 0.3989422804014327f * expf(-0.5f * zz * zz);
      float hh = zz * cdf;
      s_h[tid] = hh;
      s_z[tid] = cdf + zz * pdf;  // gelu'
      float p = wave_reduce_add(hh * hh);
      if (lane == 0) atomicAdd(&s_red[3], p);
    }
    __syncthreads();

    // ---- Phase D: pred = w2 @ h ; e = 2(pred - v); accumulate |e|^2 -------
    {
      float se = 0.0f;
#pragma unroll 1
      for (int q = 0; q < 8; ++q) {
        int row = wave * 8 + q;  // 32 waves * 8 = 256 rows
        const float* wrow = p2 + (size_t)row * kH;
        float acc = 0.0f;
#pragma unroll 4
        for (int i = lane; i < kH; i += 32) acc += wrow[i] * s_h[i];
        acc = wave_reduce_add(acc);
        if (lane == 0) {
          out[((size_t)b * kT + t) * kD + row] = acc;  // pred
          float ee = 2.0f * (acc - s_v[row]);
          s_e[row] = ee;
          se += ee * ee;
        }
      }
      if (lane == 0) atomicAdd(&s_red[4], se);
    }
    __syncthreads();

    // ---- Phase E: dz = (w2^T e) * gelu'(z); accumulate |dz|^2 -------------
    {
      float acc = 0.0f;
#pragma unroll 4
      for (int d = 0; d < kD; ++d) acc += s_e[d] * p2[(size_t)d * kH + tid];
      float dzv = acc * s_z[tid];
      s_dz[tid] = dzv;
      float p = wave_reduce_add(dzv * dzv);
      if (lane == 0) atomicAdd(&s_red[5], p);
    }
    __syncthreads();

    // ---- Phase F: gnorm via outer-product identity; clip; surprise --------
    if (tid == 0) {
      float g2 = s_red[5] * s_red[2] + s_red[4] * s_red[3];
      float gn = sqrtf(g2);
      out[(size_t)kB * kT * kD + (size_t)b * kT + t] = gn;  // surprise
      s_red[6] = fminf(1.0f / (gn + kEps), 1.0f);
    }
    __syncthreads();
    const float clip = s_red[6];

    // ---- Phase G: w2/m2 update (g2 = e  outer h), row-major, coalesced ----
    {
      const float hh = s_h[tid];
#pragma unroll 2
      for (int i = 0; i < kD; ++i) {
        int idx = i * kH + tid;
        float g = nan_to_num(s_e[i] * hh);
        float m = kMu * m2[idx] + g * clip;
        m2[idx] = m;
        p2[idx] = p2[idx] * (1.0f - kDecay) - kLR * m;
      }
    }

    // ---- Phase H: w1/m1 update in the WMMA-swizzled layout ----------------
#pragma unroll 2
    for (int i = tid; i < kW1Elems; i += 1024) {
      int j = i & 1, l = (i >> 1) & 31, ks = (i >> 6) & 63, r = i >> 12;
      int hr = r * 16 + (l & 15);
      int d = ks * 4 + ((l >> 4) << 1) + j;
      float g = nan_to_num(s_dz[hr] * s_xn[d]);
      float m = kMu * m1[i] + g * clip;
      m1[i] = m;
      p1[i] = p1[i] * (1.0f - kDecay) - kLR * m;
    }

    __threadfence();   // make param updates visible to all waves' next-step reads
    __syncthreads();
  }
}

extern "C" void kernel_launch(void* const* d_in, const int* in_sizes, int n_in,
                              void* d_out, int out_size, void* d_ws, size_t ws_size,
                              hipStream_t stream) {
  (void)in_sizes; (void)n_in; (void)out_size; (void)ws_size;
  const float* keys = (const float*)d_in[0];
  const float* values = (const float*)d_in[1];
  const float* w1 = (const float*)d_in[2];
  const float* w2 = (const float*)d_in[3];
  float* out = (float*)d_out;
  float* ws = (float*)d_ws;  // needs 8 * 4 * 262144 floats = 32 MB
  memory_rnn_kernel<<<dim3(kB), dim3(1024), 0, stream>>>(keys, values, w1, w2, out, ws);
}